// CopyMechanism_55113020342572
// MI455X (gfx1250) — compile-verified
//
#include <hip/hip_runtime.h>

typedef __attribute__((ext_vector_type(2))) float v2f;
typedef __attribute__((ext_vector_type(4))) float v4f;
typedef __attribute__((ext_vector_type(8))) float v8f;

#define B_DIM 16
#define T_DIM 128
#define H_DIM 512
#define V_DIM 32000
#define S_DIM 400
#define NROWS (B_DIM * T_DIM)   // 2048 (b,t) rows

// ---------------------------------------------------------------------------
// Phase 1: gate logits via V_WMMA_F32_16X16X4_F32 (full f32 precision).
// One wave (32 threads) computes 16 rows. A = 16x4 f32 chunk of the row data,
// B = 4x16 with the weight chunk broadcast across all N columns, so every
// column of D = A*B + C holds the same per-row dot product.
//
// A layout (ISA 7.12.2, 32-bit A 16x4): lanes 0-15 -> M=lane, K={0,1} in
// vgpr{0,1}; lanes 16-31 -> M=lane-16, K={2,3}.  B layout (4x16, row across
// lanes): lanes 0-15 -> K={0,1}, lanes 16-31 -> K={2,3}; N = lane&15 (we
// broadcast w across N, so every lane in a half just loads w[k+koff(+1)]).
// ---------------------------------------------------------------------------
__device__ inline v8f acc_dot(v8f acc, const float* __restrict__ X,
                              const float* __restrict__ W, int row, int koff) {
  const float* xr = X + (size_t)row * H_DIM;
#pragma unroll 4
  for (int k = 0; k < H_DIM; k += 4) {
    v2f a = *(const v2f*)(xr + k + koff);
    v2f b = *(const v2f*)(W + k + koff);
    // 8 args: (neg_a, A, neg_b, B, c_mod, C, reuse_a, reuse_b)
    acc = __builtin_amdgcn_wmma_f32_16x16x4_f32(false, a, false, b,
                                                (short)0, acc, false, false);
  }
  return acc;
}

__global__ __launch_bounds__(32) void pgen_wmma_kernel(
    const float* __restrict__ ctx, const float* __restrict__ hid,
    const float* __restrict__ emb, const float* __restrict__ w_h,
    const float* __restrict__ w_s, const float* __restrict__ w_x,
    const float* __restrict__ w_x_b, float* __restrict__ p_gen) {
  const int lane    = threadIdx.x;            // 0..31, full wave, EXEC all 1s
  const int m       = lane & 15;              // row within 16-row tile
  const int koff    = (lane >> 4) << 1;       // 0 for lanes 0-15, 2 for 16-31
  const int rowBase = blockIdx.x * 16;
  const int row     = rowBase + m;

  v8f acc = {};
  acc = acc_dot(acc, ctx, w_h, row, koff);
  acc = acc_dot(acc, hid, w_s, row, koff);
  acc = acc_dot(acc, emb, w_x, row, koff);

  // D layout: vgpr r, lanes 0-15 -> M=r (N=lane); lanes 16-31 -> M=8+r.
  // Column N=0 lanes (lane 0 and lane 16) hold rows base+0..7 / base+8..15.
  if (m == 0) {
    const float bias = w_x_b[0];
    const int mb = rowBase + ((lane >> 4) << 3);
#pragma unroll
    for (int r = 0; r < 8; ++r) {
      float g = acc[r] + bias;
      p_gen[mb + r] = 1.0f / (1.0f + __expf(-g));   // sigmoid
    }
  }
}

// ---------------------------------------------------------------------------
// Phase 2: out = p_gen[row] * vocab_dists  (the bandwidth phase: 524 MB).
// float4 (b128) streaming, non-temporal on both sides (no reuse; keep the
// 192MB L2 clean for everything else).
// ---------------------------------------------------------------------------
__global__ __launch_bounds__(256) void scale_vocab_kernel(
    const float* __restrict__ vocab, const float* __restrict__ p_gen,
    float* __restrict__ out) {
  const int row = blockIdx.y;                       // 0..2047 (uniform -> SGPR)
  const int i4  = blockIdx.x * 256 + threadIdx.x;   // float4 index, V/4 = 8000
  if (i4 < V_DIM / 4) {
    const float p = p_gen[row];
    const v4f* src = (const v4f*)(vocab + (size_t)row * V_DIM);
    v4f*       dst = (v4f*)(out + (size_t)row * V_DIM);
    v4f v = __builtin_nontemporal_load(src + i4);
    v4f o = p * v;
    __builtin_nontemporal_store(o, dst + i4);
  }
}

// ---------------------------------------------------------------------------
// Phase 3: scatter-add (1-p)*attn into out at src_ids, per row.
// Indices may collide within a row -> hardware f32 atomics.
// ---------------------------------------------------------------------------
__global__ __launch_bounds__(256) void scatter_add_kernel(
    const float* __restrict__ attn, const int* __restrict__ ids,
    const float* __restrict__ p_gen, float* __restrict__ out, int total) {
  const int idx = blockIdx.x * 256 + threadIdx.x;   // over 2048*400
  if (idx < total) {
    const int row = idx / S_DIM;
    const float q = 1.0f - p_gen[row];
    const int v   = ids[idx];
    unsafeAtomicAdd(out + (size_t)row * V_DIM + v, q * attn[idx]);
  }
}

// ---------------------------------------------------------------------------
// Inputs (setup_inputs order):
//  0 context_vecs [B,T,H] f32   1 hidden [B,T,H] f32   2 trg_embs [B,T,H] f32
//  3 vocab_dists [B,T,V] f32    4 attn_dists [B,T,S] f32
//  5 src_ids [B,T,S] i32        6 pad_id (unused)
//  7 w_h [1,H] f32  8 w_s [1,H] f32  9 w_x_w [1,H] f32  10 w_x_b [1] f32
// Output: [B,T,V] f32.  d_ws: p_gen (2048 floats).
// ---------------------------------------------------------------------------
extern "C" void kernel_launch(void* const* d_in, const int* in_sizes, int n_in,
                              void* d_out, int out_size, void* d_ws,
                              size_t ws_size, hipStream_t stream) {
  const float* ctx   = (const float*)d_in[0];
  const float* hid   = (const float*)d_in[1];
  const float* emb   = (const float*)d_in[2];
  const float* vocab = (const float*)d_in[3];
  const float* attn  = (const float*)d_in[4];
  const int*   ids   = (const int*)d_in[5];
  const float* w_h   = (const float*)d_in[7];
  const float* w_s   = (const float*)d_in[8];
  const float* w_x   = (const float*)d_in[9];
  const float* w_x_b = (const float*)d_in[10];
  float* out  = (float*)d_out;
  float* pgen = (float*)d_ws;   // 2048 floats of scratch

  // Phase 1: 128 waves, 16 rows each, WMMA f32 accumulation over K=1536.
  pgen_wmma_kernel<<<NROWS / 16, 32, 0, stream>>>(ctx, hid, emb, w_h, w_s,
                                                  w_x, w_x_b, pgen);

  // Phase 2: one row per blockIdx.y, float4-vectorized streaming.
  dim3 g2((V_DIM / 4 + 255) / 256, NROWS);
  scale_vocab_kernel<<<g2, 256, 0, stream>>>(vocab, pgen, out);

  // Phase 3: atomic scatter-add.
  const int total = NROWS * S_DIM;
  scatter_add_kernel<<<(total + 255) / 256, 256, 0, stream>>>(attn, ids, pgen,
                                                              out, total);
}